// TimeVaryingBiquad_78975858638864
// MI455X (gfx1250) — compile-verified
//
#include <hip/hip_runtime.h>
#include <cstdint>
#include <math.h>

// ---------------------------------------------------------------------------
// Time-varying all-pole biquad, parallelized as a 3-phase chunked linear scan.
//   B = 64 rows, T = 131072 samples.
//   Phase 1 (k1): per (row, chunk) compute affine chunk map  s_out = H*s_in + P
//   Phase 2 (k2): per row, wave32 Kogge-Stone scan over chunk maps -> chunk inits
//   Phase 3 (k3): per (row, chunk) replay recurrence with exact init, write y
// Chunk summaries / inits live in the first 6 / 2 output slots of each chunk
// (overwritten last by phase 3), so no d_ws assumptions are needed.
// CDNA5 paths: global_load_async_to_lds_b64 / global_store_async_from_lds_b64
// staging (coalesced 128B wave runs -> bank-conflict-free padded LDS tiles),
// s_wait_asynccnt, wave32 shuffle scan.
// ---------------------------------------------------------------------------

#define BROWS 64
#define TLEN  131072
#define NC    1024                 // chunks per row
#define CLEN  (TLEN / NC)          // 128 samples per chunk
#define CPB   256                  // chunks (=threads) per block
#define BPR   (NC / CPB)           // 4 blocks per row
#define SUB   32                   // samples staged per tile iteration
#define NSUB  (CLEN / SUB)         // 4
#define LSTR  34                   // LDS tile row stride (floats): 34*t mod 64 is a
                                   // permutation over 32 lanes (conflict-free) and
                                   // 34*4=136 B keeps B64 LDS writes 8B-aligned

// --- CDNA5 async global<->LDS helpers (gfx1250; tracked by ASYNCcnt) --------
__device__ __forceinline__ unsigned ldsOffset(const void* p) {
  // For generic pointers to LDS, the low 32 bits of the flat address are the
  // LDS byte offset (flat = {SHARED_BASE aperture, lds_offset}).
  return (unsigned)(uintptr_t)p;
}

__device__ __forceinline__ void asyncLoadB64(unsigned ldsOff, const float* g) {
  asm volatile("global_load_async_to_lds_b64 %0, %1, off"
               :: "v"(ldsOff), "v"((unsigned long long)(uintptr_t)g)
               : "memory");
}

__device__ __forceinline__ void asyncStoreB64(float* g, unsigned ldsOff) {
  asm volatile("global_store_async_from_lds_b64 %0, %1, off"
               :: "v"((unsigned long long)(uintptr_t)g), "v"(ldsOff)
               : "memory");
}

__device__ __forceinline__ void waitAsync0() {
  asm volatile("s_wait_asynccnt 0" ::: "memory");
}

// --- per-sample biquad denominator coefficients (matches reference math) ----
__device__ __forceinline__ void biquadCoeffs(float cmv, float rmv,
                                             float& a1, float& a2) {
  constexpr float LOG_MIN_W = -6.907755278982137f;   // ln(0.001)
  constexpr float LOG_MAX_W =  1.1447298858494002f;  // ln(pi)
  constexpr float LOG_MIN_Q = -0.3465831784519299f;  // ln(0.7071)
  constexpr float LOG_MAX_Q =  1.3862943611198906f;  // ln(4.0)
  float w = expf(LOG_MIN_W + (LOG_MAX_W - LOG_MIN_W) * cmv);
  float q = expf(LOG_MIN_Q + (LOG_MAX_Q - LOG_MIN_Q) * rmv);
  float sw, cw;
  sincosf(w, &sw, &cw);
  float alpha = sw / (2.0f * q);
  float a0inv = 1.0f / (1.0f + alpha);
  a1 = (-2.0f * cw) * a0inv;
  a2 = (1.0f - alpha) * a0inv;
}

// ---------------------------------------------------------------------------
// Phase 1: per-chunk affine state map via 3 simultaneous recurrences.
// ---------------------------------------------------------------------------
extern "C" __global__ void __launch_bounds__(CPB)
tvbq_k1_summary(const float* __restrict__ x, const float* __restrict__ cm,
                const float* __restrict__ rm, float* __restrict__ out) {
  extern __shared__ float smem[];
  float* tx = smem;
  float* tc = smem + CPB * LSTR;
  float* tr = smem + 2 * CPB * LSTR;

  const int t       = threadIdx.x;
  const int row     = blockIdx.x / BPR;
  const int cb      = (blockIdx.x % BPR) * CPB;
  const int rowBase = row * TLEN;
  const int myBase  = rowBase + (cb + t) * CLEN;

  float p1 = 0.f, p2 = 0.f;                // particular, init (0,0)
  float u1 = 1.f, u2 = 0.f;                // homogeneous, init (1,0)
  float v1 = 0.f, v2 = 1.f;                // homogeneous, init (0,1)

  for (int i = 0; i < NSUB; ++i) {
    const int s0 = i * SUB;
    // Stage [CPB chunks x SUB samples] x3 arrays: each half-wave's 16 lanes issue a
    // contiguous 128 B run; LDS dest uses padded stride via per-lane offsets.
#pragma unroll
    for (int j = 0; j < 16; ++j) {
      const int f  = j * CPB + t;
      const int ch = f >> 4;
      const int sp = (f & 15) * 2;
      const int g  = rowBase + (cb + ch) * CLEN + s0 + sp;
      const unsigned l = (unsigned)((ch * LSTR + sp) * sizeof(float));
      asyncLoadB64(ldsOffset(tx) + l, x + g);
      asyncLoadB64(ldsOffset(tc) + l, cm + g);
      asyncLoadB64(ldsOffset(tr) + l, rm + g);
    }
    waitAsync0();
    __syncthreads();
#pragma unroll 1
    for (int s = 0; s < SUB; ++s) {
      const int o = t * LSTR + s;
      float a1, a2;
      biquadCoeffs(tc[o], tr[o], a1, a2);
      float yp = tx[o] - a1 * p1 - a2 * p2;  p2 = p1;  p1 = yp;
      float yu =       - a1 * u1 - a2 * u2;  u2 = u1;  u1 = yu;
      float yv =       - a1 * v1 - a2 * v2;  v2 = v1;  v1 = yv;
    }
    __syncthreads();  // protect tiles before next iteration's async loads
  }
  // s_out = H * s_in + P, H columns from the two homogeneous runs.
  out[myBase + 0] = u1;  // H11
  out[myBase + 1] = u2;  // H21
  out[myBase + 2] = v1;  // H12
  out[myBase + 3] = v2;  // H22
  out[myBase + 4] = p1;  // P1
  out[myBase + 5] = p2;  // P2
}

// ---------------------------------------------------------------------------
// Phase 2: per-row wave32 Kogge-Stone scan over chunk affine maps.
// Combine is non-commutative:  (later) o (earlier):
//   M = Ml*Me ; v = Ml*ve + vl
// ---------------------------------------------------------------------------
extern "C" __global__ void __launch_bounds__(32)
tvbq_k2_scan(float* __restrict__ out) {
  const int row     = blockIdx.x;
  const int l       = threadIdx.x;
  const int rowBase = row * TLEN;
  float s1 = 0.f, s2 = 0.f;                // row initial state (zeros)

  for (int it = 0; it < NC / 32; ++it) {
    const int b = rowBase + (it * 32 + l) * CLEN;
    float h11 = out[b + 0], h21 = out[b + 1];
    float h12 = out[b + 2], h22 = out[b + 3];
    float p1  = out[b + 4], p2  = out[b + 5];
#pragma unroll
    for (int d = 1; d < 32; d <<= 1) {
      float o11 = __shfl_up(h11, d, 32), o21 = __shfl_up(h21, d, 32);
      float o12 = __shfl_up(h12, d, 32), o22 = __shfl_up(h22, d, 32);
      float op1 = __shfl_up(p1,  d, 32), op2 = __shfl_up(p2,  d, 32);
      if (l >= d) {
        float n11 = h11 * o11 + h12 * o21;
        float n21 = h21 * o11 + h22 * o21;
        float n12 = h11 * o12 + h12 * o22;
        float n22 = h21 * o12 + h22 * o22;
        float np1 = h11 * op1 + h12 * op2 + p1;
        float np2 = h21 * op1 + h22 * op2 + p2;
        h11 = n11; h21 = n21; h12 = n12; h22 = n22; p1 = np1; p2 = np2;
      }
    }
    // Exclusive map for this lane (identity at lane 0) -> chunk initial state.
    float e11 = __shfl_up(h11, 1, 32), e21 = __shfl_up(h21, 1, 32);
    float e12 = __shfl_up(h12, 1, 32), e22 = __shfl_up(h22, 1, 32);
    float ep1 = __shfl_up(p1,  1, 32), ep2 = __shfl_up(p2,  1, 32);
    if (l == 0) { e11 = 1.f; e21 = 0.f; e12 = 0.f; e22 = 1.f; ep1 = 0.f; ep2 = 0.f; }
    out[b + 0] = e11 * s1 + e12 * s2 + ep1;   // y[n-1] at chunk start
    out[b + 1] = e21 * s1 + e22 * s2 + ep2;   // y[n-2] at chunk start
    // Carry = inclusive map of lane 31 applied to the block-entry state.
    float c11 = __shfl(h11, 31, 32), c21 = __shfl(h21, 31, 32);
    float c12 = __shfl(h12, 31, 32), c22 = __shfl(h22, 31, 32);
    float cp1 = __shfl(p1,  31, 32), cp2 = __shfl(p2,  31, 32);
    float ns1 = c11 * s1 + c12 * s2 + cp1;
    float ns2 = c21 * s1 + c22 * s2 + cp2;
    s1 = ns1; s2 = ns2;
  }
}

// ---------------------------------------------------------------------------
// Phase 3: replay recurrence per chunk with exact init; stage output through
// LDS and write with async stores (coalesced 128 B wave runs).
// ---------------------------------------------------------------------------
extern "C" __global__ void __launch_bounds__(CPB)
tvbq_k3_apply(const float* __restrict__ x, const float* __restrict__ cm,
              const float* __restrict__ rm, float* __restrict__ out) {
  extern __shared__ float smem[];
  float* tx = smem;
  float* tc = smem + CPB * LSTR;
  float* tr = smem + 2 * CPB * LSTR;
  float* ty = smem + 3 * CPB * LSTR;

  const int t       = threadIdx.x;
  const int row     = blockIdx.x / BPR;
  const int cb      = (blockIdx.x % BPR) * CPB;
  const int rowBase = row * TLEN;
  const int myBase  = rowBase + (cb + t) * CLEN;

  // Chunk initial state written by phase 2 (consumed before any store of this
  // kernel can touch it: the post-compute barrier precedes all async stores).
  float y1 = out[myBase + 0];
  float y2 = out[myBase + 1];

  for (int i = 0; i < NSUB; ++i) {
    const int s0 = i * SUB;
#pragma unroll
    for (int j = 0; j < 16; ++j) {
      const int f  = j * CPB + t;
      const int ch = f >> 4;
      const int sp = (f & 15) * 2;
      const int g  = rowBase + (cb + ch) * CLEN + s0 + sp;
      const unsigned l = (unsigned)((ch * LSTR + sp) * sizeof(float));
      asyncLoadB64(ldsOffset(tx) + l, x + g);
      asyncLoadB64(ldsOffset(tc) + l, cm + g);
      asyncLoadB64(ldsOffset(tr) + l, rm + g);
    }
    waitAsync0();        // loads done; also drains this wave's prior stores
    __syncthreads();     // => all waves' prior stores drained before ty reuse
#pragma unroll 1
    for (int s = 0; s < SUB; ++s) {
      const int o = t * LSTR + s;
      float a1, a2;
      biquadCoeffs(tc[o], tr[o], a1, a2);
      float y = tx[o] - a1 * y1 - a2 * y2;
      y2 = y1; y1 = y;
      ty[o] = y;
    }
    __syncthreads();     // ty fully written; tiles free for next iteration
#pragma unroll
    for (int j = 0; j < 16; ++j) {
      const int f  = j * CPB + t;
      const int ch = f >> 4;
      const int sp = (f & 15) * 2;
      const int g  = rowBase + (cb + ch) * CLEN + s0 + sp;
      const unsigned l = (unsigned)((ch * LSTR + sp) * sizeof(float));
      asyncStoreB64(out + g, ldsOffset(ty) + l);
    }
  }
  // S_ENDPGM performs an implicit wait-idle, draining remaining async stores.
}

// ---------------------------------------------------------------------------
extern "C" void kernel_launch(void* const* d_in, const int* in_sizes, int n_in,
                              void* d_out, int out_size, void* d_ws, size_t ws_size,
                              hipStream_t stream) {
  (void)in_sizes; (void)n_in; (void)out_size; (void)d_ws; (void)ws_size;
  const float* x  = (const float*)d_in[0];
  const float* cm = (const float*)d_in[1];
  const float* rm = (const float*)d_in[2];
  float* out      = (float*)d_out;

  const dim3 grid(BROWS * BPR);   // 256 blocks
  const dim3 block(CPB);          // 256 threads = 8 waves (wave32)
  const size_t sh1 = (size_t)3 * CPB * LSTR * sizeof(float);  // ~102 KB
  const size_t sh3 = (size_t)4 * CPB * LSTR * sizeof(float);  // ~136 KB

  tvbq_k1_summary<<<grid, block, sh1, stream>>>(x, cm, rm, out);
  tvbq_k2_scan<<<dim3(BROWS), dim3(32), 0, stream>>>(out);
  tvbq_k3_apply<<<grid, block, sh3, stream>>>(x, cm, rm, out);
}